// kWTA_31104153158277
// MI455X (gfx1250) — compile-verified
//
#include <hip/hip_runtime.h>

// kWTA: per-row top-k threshold mask. One 256-thread workgroup per row of 4096 fp32.
// Row staged in LDS via CDNA5 async global->LDS DMA (global_load_async_to_lds_b128 +
// s_wait_asynccnt), threshold found by exact 4-pass radix select with 256-bin byte
// histograms; bin suffix-scan done entirely inside wave 0 with lane shuffles
// (no barriers), masked float4 write-out uses nontemporal b128 stores.

#define COLS 4096
#define THREADS 256
#define F4_PER_THREAD 4   // 4096 floats / 4 (float4) / 256 threads

typedef float nvec4 __attribute__((ext_vector_type(4)));  // native 4xf32 for NT store

__device__ __forceinline__ unsigned f2key(float f) {
  unsigned u = __float_as_uint(f);
  // monotone map: descending float order == descending unsigned key order
  return (u & 0x80000000u) ? ~u : (u | 0x80000000u);
}

__global__ __launch_bounds__(THREADS) void kwta_kernel(const float* __restrict__ x,
                                                       float* __restrict__ out,
                                                       int k) {
  __shared__ float    tile[COLS];      // 16 KB row staging
  __shared__ unsigned hist[256];
  __shared__ unsigned sel[2];          // [0] = winning bucket, [1] = new remaining

  const int tid  = threadIdx.x;
  const int lane = tid & 31;
  const long long row = blockIdx.x;
  const float* __restrict__ src = x + row * (long long)COLS;
  float* __restrict__ dst = out + row * (long long)COLS;

  // ---- stage row into LDS with async DMA (gfx1250) ----
  // GVS mode: saddr = 64-bit SGPR base, vaddr = 32-bit byte offset, vdst = LDS byte addr.
  unsigned lds_base = (unsigned)(size_t)(&tile[0]);
#pragma unroll
  for (int j = 0; j < 4; ++j) {
    unsigned off   = (unsigned)(tid * 16 + j * (THREADS * 16));  // 16 B per lane per issue
    unsigned laddr = lds_base + off;
    asm volatile("global_load_async_to_lds_b128 %0, %1, %2"
                 :
                 : "v"(laddr), "v"(off), "s"(src)
                 : "memory");
  }
#if __has_builtin(__builtin_amdgcn_s_wait_asynccnt)
  __builtin_amdgcn_s_wait_asynccnt(0);
#else
  asm volatile("s_wait_asynccnt 0" ::: "memory");
#endif
  __syncthreads();

  // ---- exact radix select of the k-th largest key (4 byte-passes) ----
  unsigned prefix    = 0u;
  unsigned remaining = (unsigned)k;
  const float4* __restrict__ t4 = (const float4*)tile;

#pragma unroll
  for (int pass = 0; pass < 4; ++pass) {
    const int      shift  = 24 - 8 * pass;
    const unsigned himask = (pass == 0) ? 0u : (0xFFFFFFFFu << (shift + 8));

    hist[tid] = 0u;
    __syncthreads();

#pragma unroll
    for (int j = 0; j < F4_PER_THREAD; ++j) {
      float4 v = t4[tid + j * THREADS];
      unsigned kx = f2key(v.x);
      unsigned ky = f2key(v.y);
      unsigned kz = f2key(v.z);
      unsigned kw = f2key(v.w);
      if ((kx & himask) == prefix) atomicAdd(&hist[(kx >> shift) & 255u], 1u);
      if ((ky & himask) == prefix) atomicAdd(&hist[(ky >> shift) & 255u], 1u);
      if ((kz & himask) == prefix) atomicAdd(&hist[(kz >> shift) & 255u], 1u);
      if ((kw & himask) == prefix) atomicAdd(&hist[(kw >> shift) & 255u], 1u);
    }
    __syncthreads();

    // ---- single-wave (wave 0) suffix scan over 256 bins, register-only ----
    if (tid < 32) {
      unsigned h[8];
      unsigned lsum = 0u;
#pragma unroll
      for (int i = 0; i < 8; ++i) {
        h[i] = hist[lane * 8 + i];
        lsum += h[i];
      }
      // inclusive suffix scan of per-lane sums across the wave (5 shuffle steps)
      unsigned s = lsum;
#pragma unroll
      for (int d = 1; d < 32; d <<= 1) {
        unsigned t = __shfl_down(s, d, 32);
        if (lane + d < 32) s += t;
      }
      const unsigned higher = s - lsum;  // elements in bins owned by lanes > this lane
      // walk local bins from high to low; exactly one (lane,i) straddles `remaining`
      unsigned suffix = 0u;
#pragma unroll
      for (int i = 7; i >= 0; --i) {
        const unsigned gt = higher + suffix;   // count strictly above bin lane*8+i
        suffix += h[i];
        const unsigned ge = higher + suffix;   // count at-or-above bin lane*8+i
        if (ge >= remaining && gt < remaining) {
          sel[0] = (unsigned)(lane * 8 + i);
          sel[1] = remaining - gt;
        }
      }
    }
    __syncthreads();
    prefix   |= (sel[0] << shift);
    remaining = sel[1];
    __syncthreads();
  }

  const unsigned thresh_key = prefix;  // key of the k-th largest element

  // ---- masked write-out: comp * x, exactly like the reference ----
#pragma unroll
  for (int j = 0; j < F4_PER_THREAD; ++j) {
    float4 v = t4[tid + j * THREADS];
    nvec4 o;
    o.x = ((f2key(v.x) >= thresh_key) ? 1.0f : 0.0f) * v.x;
    o.y = ((f2key(v.y) >= thresh_key) ? 1.0f : 0.0f) * v.y;
    o.z = ((f2key(v.z) >= thresh_key) ? 1.0f : 0.0f) * v.z;
    o.w = ((f2key(v.w) >= thresh_key) ? 1.0f : 0.0f) * v.w;
    __builtin_nontemporal_store(o, (nvec4*)dst + tid + j * THREADS);
  }
}

extern "C" void kernel_launch(void* const* d_in, const int* in_sizes, int n_in,
                              void* d_out, int out_size, void* d_ws, size_t ws_size,
                              hipStream_t stream) {
  (void)n_in; (void)d_ws; (void)ws_size; (void)out_size;
  const float* x = (const float*)d_in[0];
  float* out = (float*)d_out;
  const int n = in_sizes[0];
  const int rows = n / COLS;               // 8192
  const int k = (int)(0.2f * (float)COLS); // 819
  kwta_kernel<<<rows, THREADS, 0, stream>>>(x, out, k);
}